// PFNetDense_67791763800347
// MI455X (gfx1250) — compile-verified
//
#include <hip/hip_runtime.h>
#include <math.h>

// ---------------------------------------------------------------------------
// PFNetDense (LSH-binned GHConv) for MI455X / gfx1250, wave32 + f32 WMMA.
// v3.1: async global->LDS staging (GLOBAL_LOAD_ASYNC_TO_LDS_B128 / ASYNCcnt)
//       with double-buffered B panels; half-bin blocks for panel GEMMs.
//       Fix: builtin takes v4i (vector_size(16) int) pointers, AS1 / AS3.
// ---------------------------------------------------------------------------

#define BB    8
#define NN    12800
#define DM    128
#define FF    256
#define NBINS 100
#define PP    128            // BIN_SIZE
#define HR    64             // half-bin rows per block in panel GEMMs
#define NHALF 50
#define NKEY  256            // padded key range (max real key = 198)

typedef __attribute__((ext_vector_type(2))) float v2f;
typedef __attribute__((ext_vector_type(8))) float v8f;
typedef int v4i_vs __attribute__((__vector_size__(16)));

static __device__ __forceinline__ v8f wmma4(v2f a, v2f b, v8f c) {
  // D = A(16x4 f32) * B(4x16 f32) + C(16x16 f32)
  return __builtin_amdgcn_wmma_f32_16x16x4_f32(
      false, a, false, b, (short)0, c, false, false);
}

// ---- async global->LDS (CDNA5 GLOBAL_LOAD_ASYNC_TO_LDS_B128, ASYNCcnt) ----
#if defined(__has_builtin)
#if __has_builtin(__builtin_amdgcn_global_load_async_to_lds_b128) && \
    __has_builtin(__builtin_amdgcn_s_wait_asynccnt)
#define ASYNC_LDS 1
#endif
#endif

#if defined(ASYNC_LDS)
static __device__ __forceinline__ void cp16(const float* g, float* l) {
  // generic LDS address low 32 bits == LDS offset (ISA 10.2 aperture rules)
  __builtin_amdgcn_global_load_async_to_lds_b128(
      (__attribute__((address_space(1))) v4i_vs*)(unsigned long long)(g),
      (__attribute__((address_space(3))) v4i_vs*)(unsigned long long)(unsigned)(unsigned long long)(l),
      0, 0);
}
#define WAITA() __builtin_amdgcn_s_wait_asynccnt(0)
#else
static __device__ __forceinline__ void cp16(const float* g, float* l) {
  *(float4*)l = *(const float4*)g;
}
#define WAITA() ((void)0)
#endif

// ---------------------------------------------------------------------------
__global__ void k_zero_i32(int* __restrict__ p, int n) {
  int t = blockIdx.x * blockDim.x + threadIdx.x;
  if (t < n) p[t] = 0;
}

// ---------------------------------------------------------------------------
// LSH key per point: argmax over [x@cb, -x@cb] (first-max), +99 if !msk
__global__ void k_keys(const float* __restrict__ xmsg,
                       const int*   __restrict__ msk,
                       const float* __restrict__ cb,   // (DM,100), cols 0..49
                       int*         __restrict__ keys) {
  int t = blockIdx.x * blockDim.x + threadIdx.x;
  if (t >= BB * NN) return;
  const float* x = xmsg + (size_t)t * DM;
  float acc[NHALF];
#pragma unroll
  for (int k = 0; k < NHALF; k++) acc[k] = 0.f;
  for (int d = 0; d < DM; d++) {
    float xv = x[d];
    const float* crow = cb + d * 100;
#pragma unroll
    for (int k = 0; k < NHALF; k++) acc[k] = fmaf(xv, crow[k], acc[k]);
  }
  float best = acc[0];
  int bi = 0;
#pragma unroll
  for (int k = 1; k < NHALF; k++) { if (acc[k] > best) { best = acc[k]; bi = k; } }
#pragma unroll
  for (int k = 0; k < NHALF; k++) { float v = -acc[k]; if (v > best) { best = v; bi = NHALF + k; } }
  keys[t] = bi + (msk[t] ? 0 : (NBINS - 1));
}

// ---------------------------------------------------------------------------
__global__ void k_hist(const int* __restrict__ keys, int* __restrict__ hist) {
  int t = blockIdx.x * blockDim.x + threadIdx.x;
  if (t >= BB * NN) return;
  atomicAdd(&hist[(t / NN) * NKEY + keys[t]], 1);
}

__global__ void k_scan(const int* __restrict__ hist, int* __restrict__ offs) {
  int b = threadIdx.x;
  if (b < BB) {
    int acc = 0;
    for (int k = 0; k < NKEY; k++) { offs[b * NKEY + k] = acc; acc += hist[b * NKEY + k]; }
  }
}

// ---------------------------------------------------------------------------
// stable counting-sort placement: one wave32 per (batch,key), ballot ranks
__global__ void k_place(const int* __restrict__ keys,
                        const int* __restrict__ offs,
                        int*       __restrict__ flat) {
  int gw   = (blockIdx.x * blockDim.x + threadIdx.x) >> 5;
  int lane = threadIdx.x & 31;
  if (gw >= BB * NKEY) return;                 // wave-uniform
  int b = gw >> 8, k = gw & (NKEY - 1);
  int base = offs[b * NKEY + k];
  int running = 0;
  for (int n0 = 0; n0 < NN; n0 += 32) {
    int n = n0 + lane;
    bool match = (keys[b * NN + n] == k);
    unsigned mask = __builtin_amdgcn_ballot_w32(match);
    if (match) {
      int rank = __popc(mask & ((1u << lane) - 1u));
      flat[b * NN + base + running + rank] = n;
    }
    running += __popc(mask);
  }
}

// ---------------------------------------------------------------------------
// dmn = norm_i * dm * norm_q, fused: Gram (WMMA) -> dm -> rowsum -> norm.
// LDS buffer aliased: phase 1 holds Xm, phase 2 holds dm.  (~66 KB LDS)
__global__ void k_dmn(const float* __restrict__ xmsg,
                      const int*   __restrict__ msk,
                      const int*   __restrict__ flat,
                      float*       __restrict__ dmnw) {
  __shared__ float sBuf[PP * DM];   // 64 KB : Xm then dm
  __shared__ float sNa[PP];
  __shared__ float sMsk[PP];
  __shared__ float sNorm[PP];
  __shared__ int   sIdx[PP];
  int blk = blockIdx.x;             // b*100 + bin
  int b = blk / NBINS;
  int binStart = (blk % NBINS) * PP;
  int t = threadIdx.x;
  if (t < PP) {
    int idx = flat[b * NN + binStart + t];
    sIdx[t] = idx;
    sMsk[t] = msk[b * NN + idx] ? 1.0f : 0.0f;
  }
  __syncthreads();
  for (int e = t; e < PP * (DM / 4); e += 256) {        // async row gather
    int row = e >> 5, c4 = (e & 31) << 2;
    cp16(&xmsg[((size_t)b * NN + sIdx[row]) * DM + c4], &sBuf[row * DM + c4]);
  }
  WAITA();
  __syncthreads();
  if (t < PP) {
    float s = 0.f;
    const float* r = &sBuf[t * DM];
    for (int c = 0; c < DM; c++) s = fmaf(r[c], r[c], s);
    sNa[t] = s;
  }
  __syncthreads();

  int w = t >> 5, lane = t & 31, m16 = lane & 15, kh = lane >> 4;
  int tm = w;                       // tile row 0..7
  const float* arow = &sBuf[(tm * 16 + m16) * DM];
  v8f dmv[8];                       // dm tiles held in registers across alias
#pragma unroll
  for (int tn = 0; tn < 8; tn++) {
    const float* brow = &sBuf[(tn * 16 + m16) * DM];    // B = Xm^T
    v8f c = {};
    for (int k = 0; k < DM; k += 4) {
      int ka = k + 2 * kh;
      v2f a  = { arow[ka], arow[ka + 1] };
      v2f bb = { brow[ka], brow[ka + 1] };
      c = wmma4(a, bb, c);
    }
#pragma unroll
    for (int r = 0; r < 8; r++) {
      int i = tm * 16 + r + 8 * kh;
      int j = tn * 16 + m16;
      float sq = sNa[i] - 2.f * c[r] + sNa[j];
      float d  = sqrtf(fmaxf(sq, 1e-6f));
      float e  = fminf(fmaxf(__expf(-0.1f * d), 0.0f), 1.0f);  // clip [0,1]
      dmv[tn][r] = e * sMsk[i] * sMsk[j];
    }
  }
  __syncthreads();                  // all Gram reads of sBuf complete
#pragma unroll
  for (int tn = 0; tn < 8; tn++) {
#pragma unroll
    for (int r = 0; r < 8; r++) {
      sBuf[(tm * 16 + r + 8 * kh) * PP + tn * 16 + m16] = dmv[tn][r];
    }
  }
  __syncthreads();
  if (t < PP) {                     // rowsum with bank rotation
    float s = 0.f;
    for (int c = 0; c < PP; c++) { int cc = (c + t) & (PP - 1); s += fabsf(sBuf[t * PP + cc]); }
    float indeg = fminf(fmaxf(s, 0.f), 1000.f);
    sNorm[t] = (1.0f / sqrtf(indeg + 1e-6f)) * sMsk[t];
  }
  __syncthreads();
  float* dst = dmnw + (size_t)blk * (PP * PP);
  for (int e = t; e < PP * PP; e += 256) {
    int i = e >> 7, j = e & (PP - 1);
    dst[e] = sBuf[e] * sNorm[i] * sNorm[j];             // coalesced store
  }
}

// ---------------------------------------------------------------------------
// fh1_u = Xn_binned @ theta : half-bin blocks (64 rows, 128 threads),
// async double-buffered theta panels.
__global__ void k_fh1(const float* __restrict__ xnode,
                      const float* __restrict__ theta,
                      const int*   __restrict__ flat,
                      float*       __restrict__ fh1) {
  __shared__ float sXn[HR * FF];    // 64 KB
  __shared__ float sBp[2][FF * 16]; // 2 x 16 KB panels
  __shared__ int   sIdx[HR];
  int blk = blockIdx.x;             // (b*100+bin)*2 + half
  int bin2 = blk >> 1, half = blk & 1;
  int b = bin2 / NBINS;
  int rowBase = (bin2 % NBINS) * PP + half * HR;
  int t = threadIdx.x;
  if (t < HR) sIdx[t] = flat[b * NN + rowBase + t];
  __syncthreads();
  for (int e = t; e < HR * (FF / 4); e += 128) {        // async row gather
    int row = e >> 6, c4 = (e & 63) << 2;
    cp16(&xnode[((size_t)b * NN + sIdx[row]) * FF + c4], &sXn[row * FF + c4]);
  }
  // prime panel 0
  for (int e = t; e < FF * 4; e += 128) {
    int rr = e >> 2, c4 = (e & 3) << 2;
    cp16(&theta[(size_t)rr * FF + c4], &sBp[0][rr * 16 + c4]);
  }
  WAITA();
  __syncthreads();
  int w = t >> 5, lane = t & 31, m16 = lane & 15, kh = lane >> 4;
  int tm = w;                       // tile row 0..3
  const float* arow = &sXn[(tm * 16 + m16) * FF];
  float* outb = fh1 + (size_t)bin2 * (PP * FF) + (size_t)half * HR * FF;
  int cur = 0;
  for (int tn = 0; tn < 16; tn++) {
    if (tn + 1 < 16) {              // stage next panel while computing
      for (int e = t; e < FF * 4; e += 128) {
        int rr = e >> 2, c4 = (e & 3) << 2;
        cp16(&theta[(size_t)rr * FF + (tn + 1) * 16 + c4], &sBp[cur ^ 1][rr * 16 + c4]);
      }
    }
    const float* bp = sBp[cur];
    v8f c = {};
    for (int k = 0; k < FF; k += 4) {
      int ka = k + 2 * kh;
      v2f a  = { arow[ka], arow[ka + 1] };
      v2f bb = { bp[ka * 16 + m16], bp[(ka + 1) * 16 + m16] };
      c = wmma4(a, bb, c);
    }
    int col = tn * 16 + m16;
#pragma unroll
    for (int r = 0; r < 8; r++) outb[(tm * 16 + r + 8 * kh) * FF + col] = c[r];
    WAITA();
    __syncthreads();
    cur ^= 1;
  }
}

// ---------------------------------------------------------------------------
// hom = dmn @ fh1_u : half-bin blocks, async double-buffered fh1 panels.
__global__ void k_hom(const float* __restrict__ dmnw,
                      const float* __restrict__ fh1,
                      float*       __restrict__ homw) {
  __shared__ float sDm[HR * PP];    // 32 KB (64 rows of dmn)
  __shared__ float sBp[2][PP * 16]; // 2 x 8 KB panels
  int blk = blockIdx.x;
  int bin2 = blk >> 1, half = blk & 1;
  int t = threadIdx.x;
  const float* src = dmnw + (size_t)bin2 * (PP * PP) + (size_t)half * HR * PP;
  for (int e = t; e < HR * (PP / 4); e += 128) {        // async copy
    int row = e >> 5, c4 = (e & 31) << 2;
    cp16(&src[(size_t)row * PP + c4], &sDm[row * PP + c4]);
  }
  const float* fb = fh1 + (size_t)bin2 * (PP * FF);
  for (int e = t; e < PP * 4; e += 128) {               // prime panel 0
    int rr = e >> 2, c4 = (e & 3) << 2;
    cp16(&fb[(size_t)rr * FF + c4], &sBp[0][rr * 16 + c4]);
  }
  WAITA();
  __syncthreads();
  float* outb = homw + (size_t)bin2 * (PP * FF) + (size_t)half * HR * FF;
  int w = t >> 5, lane = t & 31, m16 = lane & 15, kh = lane >> 4;
  int tm = w;
  const float* arow = &sDm[(tm * 16 + m16) * PP];
  int cur = 0;
  for (int tn = 0; tn < 16; tn++) {
    if (tn + 1 < 16) {
      for (int e = t; e < PP * 4; e += 128) {
        int rr = e >> 2, c4 = (e & 3) << 2;
        cp16(&fb[(size_t)rr * FF + (tn + 1) * 16 + c4], &sBp[cur ^ 1][rr * 16 + c4]);
      }
    }
    const float* bp = sBp[cur];
    v8f c = {};
    for (int k = 0; k < PP; k += 4) {
      int ka = k + 2 * kh;
      v2f a  = { arow[ka], arow[ka + 1] };
      v2f bb = { bp[ka * 16 + m16], bp[(ka + 1) * 16 + m16] };
      c = wmma4(a, bb, c);
    }
    int col = tn * 16 + m16;
#pragma unroll
    for (int r = 0; r < 8; r++) outb[(tm * 16 + r + 8 * kh) * FF + col] = c[r];
    WAITA();
    __syncthreads();
    cur ^= 1;
  }
}

// ---------------------------------------------------------------------------
// f_het = msk*(Xn@W_h); gate = sigmoid(Xn@W_t + b_t);
// out = elu(g*hom + (1-g)*f_het)*msk, scattered via the permutation.
// Half-bin blocks; W_h/W_t panels pipelined through 2 async buffers.
__global__ void k_final(const float* __restrict__ xnode,
                        const float* __restrict__ wh,
                        const float* __restrict__ wt,
                        const float* __restrict__ bt,
                        const float* __restrict__ homw,
                        const int*   __restrict__ flat,
                        const int*   __restrict__ msk,
                        float*       __restrict__ out) {
  __shared__ float sXn[HR * FF];    // 64 KB
  __shared__ float sWp[2][FF * 16]; // 2 x 16 KB panels
  __shared__ int   sIdx[HR];
  __shared__ float sMsk[HR];
  __shared__ float sBt[FF];
  int blk = blockIdx.x;
  int bin2 = blk >> 1, half = blk & 1;
  int b = bin2 / NBINS;
  int rowBase = (bin2 % NBINS) * PP + half * HR;
  int t = threadIdx.x;
  if (t < HR) {
    int idx = flat[b * NN + rowBase + t];
    sIdx[t] = idx;
    sMsk[t] = msk[b * NN + idx] ? 1.0f : 0.0f;
  }
  sBt[t] = bt[t];
  sBt[t + 128] = bt[t + 128];
  __syncthreads();
  for (int e = t; e < HR * (FF / 4); e += 128) {        // async row gather
    int row = e >> 6, c4 = (e & 63) << 2;
    cp16(&xnode[((size_t)b * NN + sIdx[row]) * FF + c4], &sXn[row * FF + c4]);
  }
  for (int e = t; e < FF * 4; e += 128) {               // prime: W_h panel 0
    int rr = e >> 2, c4 = (e & 3) << 2;
    cp16(&wh[(size_t)rr * FF + c4], &sWp[0][rr * 16 + c4]);
  }
  WAITA();
  __syncthreads();
  int w = t >> 5, lane = t & 31, m16 = lane & 15, kh = lane >> 4;
  int tm = w;
  const float* arow = &sXn[(tm * 16 + m16) * FF];
  const float* homb = homw + (size_t)bin2 * (PP * FF) + (size_t)half * HR * FF;
  int cur = 0;
  for (int tn = 0; tn < 16; tn++) {
    int col = tn * 16 + m16;
    // stage W_t(tn) while computing f_het from W_h(tn)
    for (int e = t; e < FF * 4; e += 128) {
      int rr = e >> 2, c4 = (e & 3) << 2;
      cp16(&wt[(size_t)rr * FF + tn * 16 + c4], &sWp[cur ^ 1][rr * 16 + c4]);
    }
    const float* bp0 = sWp[cur];
    v8f ce = {};
    for (int k = 0; k < FF; k += 4) {
      int ka = k + 2 * kh;
      v2f a  = { arow[ka], arow[ka + 1] };
      v2f bb = { bp0[ka * 16 + m16], bp0[(ka + 1) * 16 + m16] };
      ce = wmma4(a, bb, ce);
    }
    WAITA();
    __syncthreads();
    cur ^= 1;
    // stage W_h(tn+1) while computing gate from W_t(tn)
    if (tn + 1 < 16) {
      for (int e = t; e < FF * 4; e += 128) {
        int rr = e >> 2, c4 = (e & 3) << 2;
        cp16(&wh[(size_t)rr * FF + (tn + 1) * 16 + c4], &sWp[cur ^ 1][rr * 16 + c4]);
      }
    }
    const float* bp1 = sWp[cur];
    v8f cg = {};
    for (int k = 0; k < FF; k += 4) {
      int ka = k + 2 * kh;
      v2f a  = { arow[ka], arow[ka + 1] };
      v2f bb = { bp1[ka * 16 + m16], bp1[(ka + 1) * 16 + m16] };
      cg = wmma4(a, bb, cg);
    }
#pragma unroll
    for (int r = 0; r < 8; r++) {
      int i = tm * 16 + r + 8 * kh;
      float homv = homb[i * FF + col];                  // 64B row segments
      float fhet = ce[r] * sMsk[i];
      float g = 1.0f / (1.0f + __expf(-(cg[r] + sBt[col])));
      float h = g * homv + (1.0f - g) * fhet;
      float o = (h > 0.0f) ? h : (__expf(h) - 1.0f);    // elu
      out[((size_t)b * NN + sIdx[i]) * FF + col] = o * sMsk[i];
    }
    WAITA();
    __syncthreads();
    cur ^= 1;
  }
}

// ---------------------------------------------------------------------------
extern "C" void kernel_launch(void* const* d_in, const int* in_sizes, int n_in,
                              void* d_out, int out_size, void* d_ws, size_t ws_size,
                              hipStream_t stream) {
  const float* x_msg  = (const float*)d_in[0];
  const float* x_node = (const float*)d_in[1];
  const int*   msk    = (const int*)  d_in[2];
  const float* cb     = (const float*)d_in[3];
  const float* W_t    = (const float*)d_in[4];
  const float* b_t    = (const float*)d_in[5];
  const float* W_h    = (const float*)d_in[6];
  const float* theta  = (const float*)d_in[7];
  float*       out    = (float*)d_out;

  char* ws = (char*)d_ws;
  const size_t SZ_KEYS = (size_t)BB * NN * sizeof(int);
  const size_t SZ_HIST = (size_t)BB * NKEY * sizeof(int);
  const size_t SZ_DMN  = (size_t)BB * NBINS * PP * PP * sizeof(float);
  const size_t SZ_FH   = (size_t)BB * NBINS * PP * FF * sizeof(float);
  size_t o = 0;
  int*   keys = (int*)  (ws + o); o += SZ_KEYS;
  int*   flat = (int*)  (ws + o); o += SZ_KEYS;
  int*   hist = (int*)  (ws + o); o += SZ_HIST;
  int*   offs = (int*)  (ws + o); o += SZ_HIST;
  float* dmnw = (float*)(ws + o); o += SZ_DMN;
  float* fh1  = (float*)(ws + o); o += SZ_FH;
  float* homw = (float*)(ws + o);

  k_zero_i32<<<(BB * NKEY + 255) / 256, 256, 0, stream>>>(hist, BB * NKEY);
  k_keys    <<<(BB * NN) / 256,        256, 0, stream>>>(x_msg, msk, cb, keys);
  k_hist    <<<(BB * NN) / 256,        256, 0, stream>>>(keys, hist);
  k_scan    <<<1, 32,                       0, stream>>>(hist, offs);
  k_place   <<<(BB * NKEY) / 8,        256, 0, stream>>>(keys, offs, flat);
  k_dmn     <<<BB * NBINS,             256, 0, stream>>>(x_msg, msk, flat, dmnw);
  k_fh1     <<<BB * NBINS * 2,         128, 0, stream>>>(x_node, theta, flat, fh1);
  k_hom     <<<BB * NBINS * 2,         128, 0, stream>>>(dmnw, fh1, homw);
  k_final   <<<BB * NBINS * 2,         128, 0, stream>>>(x_node, W_h, W_t, b_t, homw,
                                                         flat, msk, out);
}